// SGConv_90108413870524
// MI455X (gfx1250) — compile-verified
//
#include <hip/hip_runtime.h>
#include <hip/hip_bf16.h>

typedef __attribute__((ext_vector_type(2))) float v2f;
typedef __attribute__((ext_vector_type(8))) float v8f;

#define FDIM 64

// ---------------- norm pipeline ----------------

__global__ void sgc_zero_f32(float* __restrict__ p, int n) {
    int i = blockIdx.x * blockDim.x + threadIdx.x;
    if (i < n) p[i] = 0.0f;
}

__global__ void sgc_degree(const int* __restrict__ dst, float* __restrict__ deg, int E) {
    int e = blockIdx.x * blockDim.x + threadIdx.x;
    if (e < E) atomicAdd(&deg[dst[e]], 1.0f);
}

__global__ void sgc_finalize_norm(float* __restrict__ norm, int n) {
    int i = blockIdx.x * blockDim.x + threadIdx.x;
    if (i < n) norm[i] = rsqrtf(fmaxf(norm[i], 1.0f));
}

// scaled[i] = in[i] * norm[row]^(1 or 2); optionally zero acc[i] for the next scatter.
// NOTE: `in` and `acc_to_zero` may alias (read-then-overwrite of the same float4
// per thread), so no __restrict__ on them.
__global__ void sgc_scale_rows(const float* in, const float* __restrict__ norm,
                               float* __restrict__ outbuf, float* acc_to_zero,
                               int total4, int squared) {
    int i = blockIdx.x * blockDim.x + threadIdx.x;
    if (i >= total4) return;
    int row = i >> 4;                 // 16 float4 per 64-float row
    float s = norm[row];
    if (squared) s *= s;
    float4 v = ((const float4*)in)[i];
    v.x *= s; v.y *= s; v.z *= s; v.w *= s;
    ((float4*)outbuf)[i] = v;
    if (acc_to_zero) ((float4*)acc_to_zero)[i] = make_float4(0.f, 0.f, 0.f, 0.f);
}

// ---------------- edge scatter-add ----------------
// One wave per edge; lane L handles features [2L, 2L+1]. Gather float2 from the
// scaled source row (random 256B rows, 25.6MB table -> L2 resident) and
// global_atomic_add_f32 into the destination row.
__global__ void sgc_scatter_add(const int* __restrict__ src, const int* __restrict__ dst,
                                const float* __restrict__ rows, float* __restrict__ acc,
                                int E) {
    long long t = (long long)blockIdx.x * blockDim.x + threadIdx.x;
    int e = (int)(t >> 5);
    if (e >= E) return;
    int lane = (int)(t & 31);
    int s = src[e];
    int d = dst[e];
    float2 v = ((const float2*)(rows + (size_t)s * FDIM))[lane];
    float* dp = acc + (size_t)d * FDIM + lane * 2;
    atomicAdd(dp,     v.x);
    atomicAdd(dp + 1, v.y);
}

// ---------------- final projection: out = A @ W + b via V_WMMA_F32_16X16X4_F32 ----
// Block = 128 threads = 4 waves; block m handles rows [16m,16m+16), wave w handles
// output columns [16w,16w+16). K=64 swept in 16 chunks of 4.
// A 16x4 f32 layout: lanes 0-15 -> M=lane, VGPR0/1 = K0/K1; lanes 16-31 -> K2/K3.
// B 4x16 mirrors with N across lanes. C/D: VGPR r -> M = r + 8*(lane>=16), N = lane&15.
__global__ void __launch_bounds__(128)
sgc_gemm_wmma(const float* __restrict__ A, const float* __restrict__ Wm,
              const float* __restrict__ bias, float* __restrict__ out) {
    const int wave = threadIdx.x >> 5;
    const int lane = threadIdx.x & 31;
    const int half = lane >> 4;       // 0 or 1
    const int l    = lane & 15;
    const int mBase = blockIdx.x * 16;
    const int nBase = wave * 16;

    v8f acc = {};
    const float* arow = A + (size_t)(mBase + l) * FDIM;

#pragma unroll
    for (int k0 = 0; k0 < FDIM; k0 += 4) {
        int ka = k0 + 2 * half;
        v2f a;
        a.x = arow[ka];
        a.y = arow[ka + 1];
        v2f b;
        b.x = Wm[(size_t)ka * FDIM + nBase + l];
        b.y = Wm[(size_t)(ka + 1) * FDIM + nBase + l];
        // (neg_a, A, neg_b, B, c_mod, C, reuse_a, reuse_b)
        acc = __builtin_amdgcn_wmma_f32_16x16x4_f32(false, a, false, b,
                                                    (short)0, acc, false, false);
    }

    float bb = bias[nBase + l];
#pragma unroll
    for (int r = 0; r < 8; ++r) {
        int m = mBase + r + 8 * half;
        out[(size_t)m * FDIM + nBase + l] = acc[r] + bb;
    }
}

// ---------------- driver ----------------

extern "C" void kernel_launch(void* const* d_in, const int* in_sizes, int n_in,
                              void* d_out, int out_size, void* d_ws, size_t ws_size,
                              hipStream_t stream) {
    const float* feat = (const float*)d_in[0];
    const int*   src  = (const int*)d_in[1];
    const int*   dst  = (const int*)d_in[2];
    const float* Wm   = (const float*)d_in[3];
    const float* bias = (const float*)d_in[4];

    const int N = in_sizes[0] / FDIM;   // 100000 (multiple of 16)
    const int E = in_sizes[1];          // 1000000

    // ws layout: norm[N] | scaled[N*64].  acc aliases d_out (overwritten by GEMM last).
    float* norm   = (float*)d_ws;
    float* scaled = norm + N;
    float* acc    = (float*)d_out;

    const int total4 = N * (FDIM / 4);

    dim3 blk(256);
    dim3 gN((N + 255) / 256);
    dim3 gE((E + 255) / 256);
    dim3 gS((unsigned)(((long long)E * 32 + 255) / 256));
    dim3 g4((total4 + 255) / 256);

    // norm = rsqrt(max(in-degree, 1))
    sgc_zero_f32<<<gN, blk, 0, stream>>>(norm, N);
    sgc_degree<<<gE, blk, 0, stream>>>(dst, norm, E);
    sgc_finalize_norm<<<gN, blk, 0, stream>>>(norm, N);

    // hop 1
    sgc_scale_rows<<<g4, blk, 0, stream>>>(feat, norm, scaled, acc, total4, 0);
    sgc_scatter_add<<<gS, blk, 0, stream>>>(src, dst, scaled, acc, E);

    // hop 2: fold trailing norm of hop1 and leading norm of hop2 into norm^2
    sgc_scale_rows<<<g4, blk, 0, stream>>>(acc, norm, scaled, acc, total4, 1);
    sgc_scatter_add<<<gS, blk, 0, stream>>>(src, dst, scaled, acc, E);

    // trailing norm of hop2 -> scaled holds final features
    sgc_scale_rows<<<g4, blk, 0, stream>>>(acc, norm, scaled, nullptr, total4, 0);

    // d_out = scaled @ W + b (fp32 WMMA)
    sgc_gemm_wmma<<<dim3(N / 16), dim3(128), 0, stream>>>(scaled, Wm, bias, (float*)d_out);
}